// AuroraBlock_14602888806500
// MI455X (gfx1250) — compile-verified
//
#include <hip/hip_runtime.h>
#include <math.h>

// ---- problem constants (from reference) ----
namespace {
constexpr int kB = 4, kT = 1024, kDS = 1024, kN = 16, kDR = 256, kH = 4,
              kDh = 64, kDF = 1024, kNK = 3, kHID = 128;
constexpr int kPos = kB * kT;                       // 4096 positions
constexpr long long kRoutSz = (long long)kPos * kN * kDR; // 16,777,216 floats
}

typedef __attribute__((ext_vector_type(2))) float v2f;
typedef __attribute__((ext_vector_type(8))) float v8f;

// ---------------- WMMA helpers (fp32, 16x16 tiles, K-step 4) ----------------
// A row-major (M x K) in LDS; B row-major (K x N) tile (pre-offset to first col).
__device__ __forceinline__ v8f tile_gemm(const float* __restrict__ A, int lda,
                                         const float* __restrict__ Bt, int ldb,
                                         int K, int lane, v8f acc) {
  const int mn   = lane & 15;          // A row / B col for this lane
  const int koff = (lane >> 4) << 1;   // 0 or 2
  for (int k0 = 0; k0 < K; k0 += 4) {
    v2f a, b;
    a.x = A[mn * lda + k0 + koff];
    a.y = A[mn * lda + k0 + koff + 1];
    b.x = Bt[(k0 + koff) * ldb + mn];
    b.y = Bt[(k0 + koff + 1) * ldb + mn];
    acc = __builtin_amdgcn_wmma_f32_16x16x4_f32(false, a, false, b,
                                                (short)0, acc, false, false);
  }
  return acc;
}

// Variant with B supplied transposed: BT is (N x K) row-major (e.g. scores = q @ k^T).
__device__ __forceinline__ v8f tile_gemm_bt(const float* __restrict__ A, int lda,
                                            const float* __restrict__ BT, int ldbt,
                                            int K, int lane, v8f acc) {
  const int mn   = lane & 15;
  const int koff = (lane >> 4) << 1;
  for (int k0 = 0; k0 < K; k0 += 4) {
    v2f a, b;
    a.x = A[mn * lda + k0 + koff];
    a.y = A[mn * lda + k0 + koff + 1];
    b.x = BT[mn * ldbt + k0 + koff];
    b.y = BT[mn * ldbt + k0 + koff + 1];
    acc = __builtin_amdgcn_wmma_f32_16x16x4_f32(false, a, false, b,
                                                (short)0, acc, false, false);
  }
  return acc;
}

__device__ __forceinline__ float gelu_tanh(float x) {
  float x3 = x * x * x;
  return 0.5f * x * (1.0f + tanhf(0.7978845608028654f * (x + 0.044715f * x3)));
}

// ---------------- shared-memory layout (~116 KB, fits 320KB WGP LDS) ----------------
struct SM {
  float s[kDS];            // 1024
  float r[kN * kDR];       // 4096 (current slot state)
  float h[kN * kDR];       // LN output / attention concat
  float q[kN * kDR];
  float k[kN * kDR];
  float v[kN * kDR];
  float hid[kN * kDR];     // FFN hidden chunk (16 x 256)
  float a1[kN * kHID];     // allocator hidden
  float sc[kH * kN * kN];  // attention scores (4 x 16 x 16)
  float red[256];
  float dh[kHID];          // difficulty hidden
  float mu[kN];
  float rsig[kN];
  float mask[kN];
  float kw[kNK];
  float sdot;
};

__device__ __forceinline__ void layernorm_16x256(SM& sm, const float* __restrict__ src,
                                                 float* __restrict__ dst,
                                                 const float* __restrict__ gamma,
                                                 const float* __restrict__ beta, int tid) {
  if (tid < kN) {
    float m = 0.f;
    for (int i = 0; i < kDR; ++i) m += src[tid * kDR + i];
    m *= (1.0f / kDR);
    float var = 0.f;
    for (int i = 0; i < kDR; ++i) { float d = src[tid * kDR + i] - m; var += d * d; }
    var *= (1.0f / kDR);
    sm.mu[tid] = m;
    sm.rsig[tid] = rsqrtf(var + 1e-5f);
  }
  __syncthreads();
  for (int j = 0; j < 16; ++j) {
    int e = tid * 16 + j;
    int row = e >> 8, col = e & 255;
    dst[e] = (src[e] - sm.mu[row]) * sm.rsig[row] * gamma[col] + beta[col];
  }
  __syncthreads();
}

// Async global->LDS copy of 16 bytes per lane (CDNA5 ASYNCcnt path, bypasses VGPRs).
__device__ __forceinline__ void async_ld_b128(void* lds_dst, const void* gsrc) {
  unsigned lds_off = (unsigned)(size_t)lds_dst;  // generic LDS addr: low 32 bits = LDS offset
  asm volatile("global_load_async_to_lds_b128 %0, %1, off"
               :: "v"(lds_off), "v"(gsrc) : "memory");
}

// ---------------- main fused kernel: one workgroup per (b,t) position ----------------
__global__ __launch_bounds__(256)
void aurora_fused(const float* __restrict__ s_in, const float* __restrict__ r_in,
                  const float* __restrict__ Wd1, const float* __restrict__ bd1,
                  const float* __restrict__ Wd2, const float* __restrict__ bd2,
                  const float* __restrict__ Wa1, const float* __restrict__ ba1,
                  const float* __restrict__ Wa2, const float* __restrict__ ba2,
                  const float* __restrict__ Ws,
                  const float* __restrict__ ln1_s, const float* __restrict__ ln1_b,
                  const float* __restrict__ Wq, const float* __restrict__ Wk,
                  const float* __restrict__ Wv, const float* __restrict__ Wo,
                  const float* __restrict__ ln2_s, const float* __restrict__ ln2_b,
                  const float* __restrict__ Wf1, const float* __restrict__ bf1,
                  const float* __restrict__ Wf2, const float* __restrict__ bf2,
                  float* __restrict__ out, float* __restrict__ accum) {
  __shared__ SM sm;
  const int tid  = threadIdx.x;
  const int lane = tid & 31;
  const int wid  = tid >> 5;
  const int p    = blockIdx.x;

  const float* sP = s_in + (size_t)p * kDS;
  const float* rP = r_in + (size_t)p * kN * kDR;

  // weights are hot in L2 across the whole grid; hint the streams
  __builtin_prefetch(Wq, 0, 1);
  __builtin_prefetch(Wf1, 0, 1);

  // ---- stage s (4KB) and r (16KB) straight into LDS via async b128 copies ----
  async_ld_b128(&sm.s[tid * 4], sP + tid * 4);          // 256 lanes x 16B = 4KB
#pragma unroll
  for (int j = 0; j < 4; ++j) {                          // 4 x 4KB = 16KB
    int c = tid + 256 * j;
    async_ld_b128(&sm.r[c * 4], rP + c * 4);
  }
  asm volatile("s_wait_asynccnt 0x0" ::: "memory");
  __syncthreads();

  // ---- DifficultyEstimator: relu(s@Wd1+bd1)@Wd2+bd2 -> softmax(3) ----
  if (tid < kHID) {
    float a = bd1[tid];
    for (int i = 0; i < kDS; ++i) a += sm.s[i] * Wd1[i * kHID + tid];
    sm.dh[tid] = fmaxf(a, 0.f);
  }
  { // sdot = s @ Ws (scalar) — tree reduce
    float part = 0.f;
    for (int i = tid * 4; i < tid * 4 + 4; ++i) part += sm.s[i] * Ws[i];
    sm.red[tid] = part;
  }
  __syncthreads();
  for (int off = 128; off > 0; off >>= 1) {
    if (tid < off) sm.red[tid] += sm.red[tid + off];
    __syncthreads();
  }
  if (tid < kNK) {
    float a = bd2[tid];
    for (int i = 0; i < kHID; ++i) a += sm.dh[i] * Wd2[i * kNK + tid];
    sm.kw[tid] = a;  // logits for now
  }
  if (tid == 0) sm.sdot = sm.red[0];
  __syncthreads();
  if (tid == 0) {
    float m = fmaxf(sm.kw[0], fmaxf(sm.kw[1], sm.kw[2]));
    float e0 = expf(sm.kw[0] - m), e1 = expf(sm.kw[1] - m), e2 = expf(sm.kw[2] - m);
    float inv = 1.f / (e0 + e1 + e2);
    float p0 = e0 * inv, p1 = e1 * inv, p2 = e2 * inv;
    sm.kw[0] = p0; sm.kw[1] = p1; sm.kw[2] = p2;
    atomicAdd(&accum[0], p0);
    atomicAdd(&accum[1], p1);
    atomicAdd(&accum[2], p2);
    atomicAdd(&accum[3], p0 * 1.f + p1 * 2.f + p2 * 4.f);  // for mean_k
  }

  // ---- SlotAllocator: sigmoid(relu(r@Wa1+ba1)@Wa2 + ba2 + s@Ws) ----
  for (int qx = tid; qx < kN * kHID; qx += 256) {
    int n = qx >> 7, j = qx & 127;
    float a = ba1[j];
    for (int i = 0; i < kDR; ++i) a += sm.r[n * kDR + i] * Wa1[i * kHID + j];
    sm.a1[qx] = fmaxf(a, 0.f);
  }
  __syncthreads();
  if (tid < kN) {
    float a = ba2[0];
    for (int j = 0; j < kHID; ++j) a += sm.a1[tid * kHID + j] * Wa2[j];
    float z = a + sm.sdot;
    sm.mask[tid] = 1.f / (1.f + expf(-z));
  }
  __syncthreads();

  // per-thread candidate accumulator (16 contiguous elements)
  float racc[16];
#pragma unroll
  for (int j = 0; j < 16; ++j) racc[j] = 0.f;

  // ---- 4 reasoning steps ----
  for (int step = 1; step <= 4; ++step) {
    // LN1: r -> h
    layernorm_16x256(sm, sm.r, sm.h, ln1_s, ln1_b, tid);

    // q/k/v GEMMs: 48 16x16 output tiles, K=256
    for (int tt = wid; tt < 48; tt += 8) {
      int mat = tt >> 4, nt = tt & 15;
      const float* W = (mat == 0 ? Wq : (mat == 1 ? Wk : Wv)) + nt * 16;
      float* dst = (mat == 0 ? sm.q : (mat == 1 ? sm.k : sm.v));
      v8f acc = {};
      acc = tile_gemm(sm.h, kDR, W, kDR, kDR, lane, acc);
      int col = nt * 16 + (lane & 15);
      int rbase = (lane >> 4) << 3;
#pragma unroll
      for (int i = 0; i < 8; ++i) dst[(rbase + i) * kDR + col] = acc[i];
    }
    __syncthreads();

    // scores[h] = q_h @ k_h^T / 8 (waves 0..3, one head each)
    if (wid < kH) {
      int hh = wid;
      v8f acc = {};
      acc = tile_gemm_bt(sm.q + hh * kDh, kDR, sm.k + hh * kDh, kDR, kDh, lane, acc);
      int col = lane & 15;
      int rbase = (lane >> 4) << 3;
#pragma unroll
      for (int i = 0; i < 8; ++i)
        sm.sc[hh * 256 + (rbase + i) * 16 + col] = acc[i] * 0.125f;
    }
    __syncthreads();

    // softmax over last axis (16) per (head, slot)
    if (tid < kH * kN) {
      float* row = sm.sc + (tid >> 4) * 256 + (tid & 15) * 16;
      float mx = row[0];
      for (int m = 1; m < 16; ++m) mx = fmaxf(mx, row[m]);
      float ssum = 0.f;
      for (int m = 0; m < 16; ++m) { row[m] = expf(row[m] - mx); ssum += row[m]; }
      float inv = 1.f / ssum;
      for (int m = 0; m < 16; ++m) row[m] *= inv;
    }
    __syncthreads();

    // attn @ v -> h (concat heads): 16 tiles, K=16
    for (int tt = wid; tt < 16; tt += 8) {
      int hh = tt >> 2, dt = tt & 3;
      v8f acc = {};
      acc = tile_gemm(sm.sc + hh * 256, 16, sm.v + hh * kDh + dt * 16, kDR, 16, lane, acc);
      int col = hh * kDh + dt * 16 + (lane & 15);
      int rbase = (lane >> 4) << 3;
#pragma unroll
      for (int i = 0; i < 8; ++i) sm.h[(rbase + i) * kDR + col] = acc[i];
    }
    __syncthreads();

    // o = h @ Wo ; r += mask * o
    for (int nt = wid; nt < 16; nt += 8) {
      v8f acc = {};
      acc = tile_gemm(sm.h, kDR, Wo + nt * 16, kDR, kDR, lane, acc);
      int col = nt * 16 + (lane & 15);
      int rbase = (lane >> 4) << 3;
#pragma unroll
      for (int i = 0; i < 8; ++i) {
        int row = rbase + i;
        sm.r[row * kDR + col] += sm.mask[row] * acc[i];
      }
    }
    __syncthreads();

    // LN2: r -> h
    layernorm_16x256(sm, sm.r, sm.h, ln2_s, ln2_b, tid);

    // FFN in 4 K-chunks of 256: hid = gelu(h@Wf1+bf1); r += mask*(hid@Wf2 + bf2)
    for (int c = 0; c < 4; ++c) {
      for (int nt = wid; nt < 16; nt += 8) {
        v8f acc = {};
        acc = tile_gemm(sm.h, kDR, Wf1 + c * 256 + nt * 16, kDF, kDR, lane, acc);
        int col = nt * 16 + (lane & 15);
        int rbase = (lane >> 4) << 3;
#pragma unroll
        for (int i = 0; i < 8; ++i) {
          float x = acc[i] + bf1[c * 256 + col];
          sm.hid[(rbase + i) * kDR + col] = gelu_tanh(x);
        }
      }
      __syncthreads();
      for (int nt = wid; nt < 16; nt += 8) {
        v8f acc = {};
        acc = tile_gemm(sm.hid, kDR, Wf2 + (size_t)c * 256 * kDR + nt * 16, kDR,
                        kDR, lane, acc);
        int col = nt * 16 + (lane & 15);
        int rbase = (lane >> 4) << 3;
#pragma unroll
        for (int i = 0; i < 8; ++i) {
          int row = rbase + i;
          float val = acc[i] + (c == 0 ? bf2[col] : 0.f);
          sm.r[row * kDR + col] += sm.mask[row] * val;
        }
      }
      __syncthreads();
    }

    // keep candidates at k in {1,2,4}
    int widx = (step == 1) ? 0 : (step == 2) ? 1 : (step == 4) ? 2 : -1;
    if (widx >= 0) {
      float w = sm.kw[widx];
#pragma unroll
      for (int j = 0; j < 16; ++j) racc[j] += w * sm.r[tid * 16 + j];
    }
    __syncthreads();
  }

  // write r_out
  float* outP = out + (size_t)p * kN * kDR;
#pragma unroll
  for (int j = 0; j < 16; ++j) outP[tid * 16 + j] = racc[j];
}

// ---------------- tiny init / finalize kernels (graph-safe) ----------------
__global__ void init_accum(float* accum) {
  if (threadIdx.x < 4 && blockIdx.x == 0) accum[threadIdx.x] = 0.f;
}

__global__ void finalize_scalars(const float* __restrict__ accum, float* __restrict__ out) {
  if (threadIdx.x == 0 && blockIdx.x == 0) {
    float inv = 1.f / (float)kPos;
    float m0 = accum[0] * inv, m1 = accum[1] * inv, m2 = accum[2] * inv;
    const float u = 1.f / 3.f;
    out[kRoutSz]     = ((m0 - u) * (m0 - u) + (m1 - u) * (m1 - u) + (m2 - u) * (m2 - u)) / 3.f;
    out[kRoutSz + 1] = accum[3] * inv;  // mean_k
  }
}

extern "C" void kernel_launch(void* const* d_in, const int* in_sizes, int n_in,
                              void* d_out, int out_size, void* d_ws, size_t ws_size,
                              hipStream_t stream) {
  const float* s    = (const float*)d_in[0];
  const float* r    = (const float*)d_in[1];
  const float* Wd1  = (const float*)d_in[2];
  const float* bd1  = (const float*)d_in[3];
  const float* Wd2  = (const float*)d_in[4];
  const float* bd2  = (const float*)d_in[5];
  const float* Wa1  = (const float*)d_in[6];
  const float* ba1  = (const float*)d_in[7];
  const float* Wa2  = (const float*)d_in[8];
  const float* ba2  = (const float*)d_in[9];
  const float* Ws   = (const float*)d_in[10];
  const float* ln1s = (const float*)d_in[11];
  const float* ln1b = (const float*)d_in[12];
  const float* Wq   = (const float*)d_in[13];
  const float* Wk   = (const float*)d_in[14];
  const float* Wv   = (const float*)d_in[15];
  const float* Wo   = (const float*)d_in[16];
  const float* ln2s = (const float*)d_in[17];
  const float* ln2b = (const float*)d_in[18];
  const float* Wf1  = (const float*)d_in[19];
  const float* bf1  = (const float*)d_in[20];
  const float* Wf2  = (const float*)d_in[21];
  const float* bf2  = (const float*)d_in[22];
  float* out   = (float*)d_out;
  float* accum = (float*)d_ws;  // 4 floats of scratch for the scalar reductions

  init_accum<<<1, 32, 0, stream>>>(accum);
  aurora_fused<<<kPos, 256, 0, stream>>>(s, r, Wd1, bd1, Wd2, bd2, Wa1, ba1, Wa2,
                                         ba2, Ws, ln1s, ln1b, Wq, Wk, Wv, Wo, ln2s,
                                         ln2b, Wf1, bf1, Wf2, bf2, out, accum);
  finalize_scalars<<<1, 32, 0, stream>>>(accum, out);
}